// GraphConvolution_29721173688343
// MI455X (gfx1250) — compile-verified
//
#include <hip/hip_runtime.h>
#include <hip/hip_bf16.h>

typedef __attribute__((ext_vector_type(2))) float v2f;
typedef __attribute__((ext_vector_type(8))) float v8f;

#define D 128  // D_IN == D_OUT == 128

// ---------------------------------------------------------------------------
// Kernel 1: out[n, j] = bias[j]   (atomics accumulate on top of this)
// ---------------------------------------------------------------------------
__global__ void gc_init_out(float* __restrict__ out,
                            const float* __restrict__ bias,
                            int total) {
    int i = blockIdx.x * blockDim.x + threadIdx.x;
    if (i < total) out[i] = bias[i & (D - 1)];
}

// ---------------------------------------------------------------------------
// Kernel 2: support = input @ weight via V_WMMA_F32_16X16X4_F32 (fp32 WMMA).
// Grid: M/16 blocks of 256 threads (8 waves). Wave w owns column tile w.
// K loop: 32 steps of 16x16x4.
// ---------------------------------------------------------------------------
__global__ void gc_gemm_wmma(const float* __restrict__ A,   // [M, 128]
                             const float* __restrict__ B,   // [128, 128]
                             float* __restrict__ C,         // [M, 128]
                             int M) {
    const int wave  = threadIdx.x >> 5;      // 0..7 -> N tile
    const int lane  = threadIdx.x & 31;
    const int l15   = lane & 15;
    const int khalf = (lane >> 4) * 2;       // 0 (lanes 0-15) or 2 (lanes 16-31)
    const int row0  = blockIdx.x * 16;
    const int col0  = wave * 16;

    // A fragment base: lane l15 holds row (row0+l15), K = khalf, khalf+1
    const float* Arow = A + (size_t)(row0 + l15) * D;
    // B fragment base: lane l15 holds col (col0+l15), K = khalf, khalf+1
    const float* Bcol = B + (col0 + l15);

    v8f c = {};
#pragma unroll
    for (int k0 = 0; k0 < D; k0 += 4) {
        v2f a, b;
        a.x = Arow[k0 + khalf];
        a.y = Arow[k0 + khalf + 1];
        b.x = Bcol[(size_t)(k0 + khalf) * D];
        b.y = Bcol[(size_t)(k0 + khalf + 1) * D];
        // (neg_a, A, neg_b, B, c_mod, C, reuse_a, reuse_b)
        c = __builtin_amdgcn_wmma_f32_16x16x4_f32(
            false, a, false, b, (short)0, c, false, false);
    }

    // C/D layout: VGPR r -> M = r + 8*(lane>=16), N = lane&15
    const int mrow = row0 + (lane >> 4) * 8;
    const int ncol = col0 + l15;
#pragma unroll
    for (int r = 0; r < 8; ++r)
        C[(size_t)(mrow + r) * D + ncol] = c[r];
}

// ---------------------------------------------------------------------------
// Kernel 3: edge-parallel scatter of (A + A^T) @ support with L2 atomics.
// One wave per edge; lane l handles float4 at offset 4*l (32*4 = 128).
// ---------------------------------------------------------------------------
__global__ void gc_edge_scatter(const float* __restrict__ support,
                                float* __restrict__ out,
                                const int* __restrict__ edge_index, // [2, E]
                                const int* __restrict__ edge_rel,   // [E]
                                const float* __restrict__ alpha,    // [N_REL+1]
                                int E) {
    const int wid  = blockIdx.x * (blockDim.x >> 5) + (threadIdx.x >> 5);
    const int lane = threadIdx.x & 31;
    if (wid >= E) return;

    const int rel = edge_rel[wid];
    const float a = (rel != 0) ? alpha[rel] : 0.0f;  // padding row -> 0
    if (a == 0.0f) return;                           // adds exactly zero

    const int row = edge_index[wid];
    const int col = edge_index[E + wid];

    const float4 vc = ((const float4*)(support + (size_t)col * D))[lane];
    const float4 vr = ((const float4*)(support + (size_t)row * D))[lane];

    float* orow = out + (size_t)row * D + lane * 4;
    float* ocol = out + (size_t)col * D + lane * 4;

    atomicAdd(orow + 0, a * vc.x);
    atomicAdd(orow + 1, a * vc.y);
    atomicAdd(orow + 2, a * vc.z);
    atomicAdd(orow + 3, a * vc.w);

    atomicAdd(ocol + 0, a * vr.x);
    atomicAdd(ocol + 1, a * vr.y);
    atomicAdd(ocol + 2, a * vr.z);
    atomicAdd(ocol + 3, a * vr.w);
}

// ---------------------------------------------------------------------------
extern "C" void kernel_launch(void* const* d_in, const int* in_sizes, int n_in,
                              void* d_out, int out_size, void* d_ws, size_t ws_size,
                              hipStream_t stream) {
    const float* input  = (const float*)d_in[0];  // [N, 128]
    const float* weight = (const float*)d_in[1];  // [128, 128]
    const float* bias   = (const float*)d_in[2];  // [128]
    const float* alpha  = (const float*)d_in[3];  // [N_REL+1, 1]
    const int*   eidx   = (const int*)d_in[4];    // [2, E]
    const int*   erel   = (const int*)d_in[5];    // [E]

    const int N = in_sizes[0] / D;   // 100000 (multiple of 16)
    const int E = in_sizes[5];       // 1600000

    float* support = (float*)d_ws;   // [N, 128] scratch
    float* out     = (float*)d_out;  // [N, 128]

    // 1) out = bias (broadcast)
    {
        const int total = N * D;
        gc_init_out<<<(total + 255) / 256, 256, 0, stream>>>(out, bias, total);
    }

    // 2) support = input @ weight   (fp32 WMMA, one block = 16 rows x 128 cols)
    gc_gemm_wmma<<<N / 16, 256, 0, stream>>>(input, weight, support, N);

    // 3) out += segment scatter of both directions (L2-resident atomics)
    {
        const int waves_per_block = 8;             // 256 threads
        const int blocks = (E + waves_per_block - 1) / waves_per_block;
        gc_edge_scatter<<<blocks, 256, 0, stream>>>(support, out, eidx, erel,
                                                    alpha, E);
    }
}